// GRCU_61658550501402
// MI455X (gfx1250) — compile-verified
//
#include <hip/hip_runtime.h>
#include <hip/hip_bf16.h>
#include <math.h>

// Problem dims (fixed by the reference).
#define T_ 16
#define N_ 4096
#define R_ 256
#define C_ 32

typedef float v2f __attribute__((ext_vector_type(2)));
typedef float v4f __attribute__((ext_vector_type(4)));
typedef float v8f __attribute__((ext_vector_type(8)));

__device__ __forceinline__ v8f zero8() {
  v8f z;
  #pragma unroll
  for (int i = 0; i < 8; ++i) z[i] = 0.0f;
  return z;
}

// D = A(16x4 f32) * B(4x16 f32) + C(16x16 f32), wave32.
// 8-arg form: (neg_a, A, neg_b, B, c_mod, C, reuse_a, reuse_b)
__device__ __forceinline__ v8f wmma4(v2f a, v2f b, v8f c) {
  return __builtin_amdgcn_wmma_f32_16x16x4_f32(false, a, false, b, (short)0, c,
                                               false, false);
}

// Fast gate nonlinearities: v_exp_f32 + v_rcp_f32, saturating correctly at
// +/-inf (no NaN paths), avoiding the IEEE v_div_scale/v_div_fmas expansion.
__device__ __forceinline__ float sigmoidf_(float x) {
  return __builtin_amdgcn_rcpf(1.0f + __expf(-x));
}
__device__ __forceinline__ float tanhf_(float x) {
  // tanh(x) = 1 - 2/(e^{2x}+1); e->inf gives 1, e->0 gives -1.
  const float e = __expf(2.0f * x);
  return 1.0f - 2.0f * __builtin_amdgcn_rcpf(e + 1.0f);
}

// ---------------------------------------------------------------------------
// Kernel 1: evolve Q through T GRU steps. Columns of Q are independent, so
// each of the 2 blocks owns a 16-column slice for the entire recurrence.
// Per step: pass1 accumulates (Wz+Uz)@Q, (Wr+Ur)@Q, Wh@Q with f32 WMMA;
// then r = sigmoid(.), Rs = r*Q in LDS; pass2 adds Uh@Rs; gates; update.
// ---------------------------------------------------------------------------
__global__ __launch_bounds__(256) void grcu_qseq(
    const float* __restrict__ Q0, const float* __restrict__ Wz,
    const float* __restrict__ Uz, const float* __restrict__ Wr,
    const float* __restrict__ Ur, const float* __restrict__ Wh,
    const float* __restrict__ Uh, float* __restrict__ Qseq) {
  __shared__ float Qs[R_ * 17];  // stride 17: conflict-free B-frag reads
  __shared__ float Rs[R_ * 17];

  const int tid = threadIdx.x;
  const int lane = tid & 31, wave = tid >> 5;
  const int lrow = lane & 15;
  const int khalf = (lane >> 4) * 2;  // 0 for lanes 0-15, 2 for lanes 16-31
  const int cb = blockIdx.x;          // column block: cols [16*cb, 16*cb+16)

  for (int idx = tid; idx < R_ * 16; idx += 256) {
    const int m = idx >> 4, n = idx & 15;
    Qs[m * 17 + n] = Q0[m * C_ + cb * 16 + n];
  }
  __syncthreads();

  const int m0a = wave * 32;       // row-tile 0 of this wave
  const int m0b = wave * 32 + 16;  // row-tile 1 of this wave

  for (int t = 0; t < T_; ++t) {
    v8f az0 = zero8(), az1 = zero8();
    v8f ar0 = zero8(), ar1 = zero8();
    v8f ah0 = zero8(), ah1 = zero8();

    // Pass 1: K over R, B-frag from Qs.
    for (int k = 0; k < R_; k += 4) {
      v2f b;
      b.x = Qs[(k + khalf) * 17 + lrow];
      b.y = Qs[(k + khalf + 1) * 17 + lrow];
      const int ra = (m0a + lrow) * R_ + k + khalf;
      const int rb = (m0b + lrow) * R_ + k + khalf;
      {
        v2f wz = *(const v2f*)(Wz + ra), uz = *(const v2f*)(Uz + ra);
        v2f wr = *(const v2f*)(Wr + ra), ur = *(const v2f*)(Ur + ra);
        v2f wh = *(const v2f*)(Wh + ra);
        az0 = wmma4(wz + uz, b, az0);
        ar0 = wmma4(wr + ur, b, ar0);
        ah0 = wmma4(wh, b, ah0);
      }
      {
        v2f wz = *(const v2f*)(Wz + rb), uz = *(const v2f*)(Uz + rb);
        v2f wr = *(const v2f*)(Wr + rb), ur = *(const v2f*)(Ur + rb);
        v2f wh = *(const v2f*)(Wh + rb);
        az1 = wmma4(wz + uz, b, az1);
        ar1 = wmma4(wr + ur, b, ar1);
        ah1 = wmma4(wh, b, ah1);
      }
    }

    // Gates z, r; stage r*Q into Rs (each wave writes only its own rows).
    v8f z0, z1;
    #pragma unroll
    for (int v = 0; v < 8; ++v) {
      z0[v] = sigmoidf_(az0[v]);
      z1[v] = sigmoidf_(az1[v]);
      const float r0 = sigmoidf_(ar0[v]);
      const float r1 = sigmoidf_(ar1[v]);
      const int mo = v + ((lane >> 4) << 3);
      const int ia = (m0a + mo) * 17 + lrow;
      const int ib = (m0b + mo) * 17 + lrow;
      Rs[ia] = r0 * Qs[ia];
      Rs[ib] = r1 * Qs[ib];
    }
    __syncthreads();

    // Pass 2: ah += Uh @ (r*Q).
    for (int k = 0; k < R_; k += 4) {
      v2f b;
      b.x = Rs[(k + khalf) * 17 + lrow];
      b.y = Rs[(k + khalf + 1) * 17 + lrow];
      const int ra = (m0a + lrow) * R_ + k + khalf;
      const int rb = (m0b + lrow) * R_ + k + khalf;
      ah0 = wmma4(*(const v2f*)(Uh + ra), b, ah0);
      ah1 = wmma4(*(const v2f*)(Uh + rb), b, ah1);
    }

    // Q_new = (1-z)*Q + z*tanh(ah); write back to Qs and to Qseq[t].
    #pragma unroll
    for (int v = 0; v < 8; ++v) {
      const int mo = v + ((lane >> 4) << 3);
      const int ia = (m0a + mo) * 17 + lrow;
      const int ib = (m0b + mo) * 17 + lrow;
      const float h0 = tanhf_(ah0[v]);
      const float h1 = tanhf_(ah1[v]);
      const float q0 = (1.0f - z0[v]) * Qs[ia] + z0[v] * h0;
      const float q1 = (1.0f - z1[v]) * Qs[ib] + z1[v] * h1;
      Qs[ia] = q0;
      Qs[ib] = q1;
      Qseq[(size_t)t * R_ * C_ + (m0a + mo) * C_ + cb * 16 + lrow] = q0;
      Qseq[(size_t)t * R_ * C_ + (m0b + mo) * C_ + cb * 16 + lrow] = q1;
    }
    __syncthreads();
  }
}

// ---------------------------------------------------------------------------
// Kernel 2: Y_t = X_t @ Qseq[t]  (4096x256 @ 256x32), written directly in
// interleaved WMMA B-fragment layout so kernel 3 loads BOTH column fragments
// of a k-step with a single contiguous global_load_b128 per lane:
//   Ysw[((t*1024 + kb)*32 + lane)*4 + {0,1,2,3}]
//     = { Yb0.x, Yb0.y, Yb1.x, Yb1.y }
// where for lane<16: Yb_c = {Y[4kb+0][16c+l], Y[4kb+1][16c+l]},
// and for lane>=16 rows 4kb+2 / 4kb+3 (matching the A-frag khalf convention).
// ---------------------------------------------------------------------------
__global__ __launch_bounds__(256) void xq_kernel(const float* __restrict__ X,
                                                 const float* __restrict__ Qseq,
                                                 float* __restrict__ Ysw) {
  __shared__ float Qs[R_ * 33];
  __shared__ float Ysh[8 * 16 * 33];
  const int tid = threadIdx.x, lane = tid & 31, wave = tid >> 5;
  const int lrow = lane & 15, khalf = (lane >> 4) * 2;
  const int t = blockIdx.x >> 5;   // 32 row-blocks per timestep
  const int rb = blockIdx.x & 31;

  for (int idx = tid; idx < R_ * C_; idx += 256)
    Qs[(idx >> 5) * 33 + (idx & 31)] = Qseq[(size_t)t * R_ * C_ + idx];
  __syncthreads();

  const int m0 = rb * 128 + wave * 16;
  const float* Xrow = X + ((size_t)t * N_ + m0 + lrow) * R_;
  v8f a0 = zero8(), a1 = zero8();

  for (int k = 0; k < R_; k += 4) {
    v2f a = *(const v2f*)(Xrow + k + khalf);
    const int q = (k + khalf) * 33 + lrow;
    v2f b0, b1;
    b0.x = Qs[q];      b0.y = Qs[q + 33];
    b1.x = Qs[q + 16]; b1.y = Qs[q + 49];
    a0 = wmma4(a, b0, a0);
    a1 = wmma4(a, b1, a1);
  }

  // Bounce this wave's 16x32 tile through LDS to emit B-fragment layout.
  float* Yw = Ysh + wave * 16 * 33;
  #pragma unroll
  for (int v = 0; v < 8; ++v) {
    const int mo = v + ((lane >> 4) << 3);
    Yw[mo * 33 + lrow] = a0[v];
    Yw[mo * 33 + 16 + lrow] = a1[v];
  }
  __syncthreads();

  #pragma unroll
  for (int q4 = 0; q4 < 4; ++q4) {
    const int rrow = 4 * q4 + khalf;
    v4f f;
    f.x = Yw[rrow * 33 + lrow];
    f.y = Yw[(rrow + 1) * 33 + lrow];
    f.z = Yw[rrow * 33 + 16 + lrow];
    f.w = Yw[(rrow + 1) * 33 + 16 + lrow];
    const size_t kb = (size_t)(m0 >> 2) + q4;
    *(v4f*)(Ysw + (((size_t)t * 1024 + kb) * 32 + lane) * 4) = f;
  }
}

// ---------------------------------------------------------------------------
// Kernel 3: out_t = relu(A_t @ Y_t). HBM-bound: streams 1.07 GB of A exactly
// once (non-temporal b64), one b128 per lane fetches both L2-resident Y
// fragments, two f32 WMMAs per k-step, fused ReLU on store.
// ---------------------------------------------------------------------------
__global__ __launch_bounds__(256) void ay_kernel(const float* __restrict__ A,
                                                 const float* __restrict__ Ysw,
                                                 float* __restrict__ out) {
  const int tid = threadIdx.x, lane = tid & 31, wave = tid >> 5;
  const int lrow = lane & 15, khalf = (lane >> 4) * 2;
  const int t = blockIdx.x >> 5;
  const int rb = blockIdx.x & 31;
  const int m0 = rb * 128 + wave * 16;

  const float* Arow = A + ((size_t)t * N_ + m0 + lrow) * N_;
  const float* Yt = Ysw + (size_t)t * 1024 * 128;
  v8f a0 = zero8(), a1 = zero8();

  #pragma unroll 8
  for (int kb = 0; kb < 1024; ++kb) {
    v2f a = __builtin_nontemporal_load((const v2f*)(Arow + 4 * kb + khalf));
    v4f bb = *(const v4f*)(Yt + (size_t)kb * 128 + lane * 4);
    v2f b0, b1;
    b0.x = bb.x; b0.y = bb.y;
    b1.x = bb.z; b1.y = bb.w;
    a0 = wmma4(a, b0, a0);
    a1 = wmma4(a, b1, a1);
  }

  float* orow = out + (size_t)t * N_ * C_;
  #pragma unroll
  for (int v = 0; v < 8; ++v) {
    const int m = m0 + v + ((lane >> 4) << 3);
    const float r0 = a0[v], r1 = a1[v];
    orow[m * C_ + lrow] = r0 > 0.0f ? r0 : 0.0f;
    orow[m * C_ + 16 + lrow] = r1 > 0.0f ? r1 : 0.0f;
  }
}

extern "C" void kernel_launch(void* const* d_in, const int* in_sizes, int n_in,
                              void* d_out, int out_size, void* d_ws,
                              size_t ws_size, hipStream_t stream) {
  (void)in_sizes; (void)n_in; (void)out_size; (void)ws_size;
  const float* A  = (const float*)d_in[0];  // [1,T,N,N]
  const float* X  = (const float*)d_in[1];  // [T,N,R]
  const float* Q0 = (const float*)d_in[2];  // [R,C]
  const float* Wz = (const float*)d_in[3];
  const float* Uz = (const float*)d_in[4];
  const float* Wr = (const float*)d_in[5];
  const float* Ur = (const float*)d_in[6];
  const float* Wh = (const float*)d_in[7];
  const float* Uh = (const float*)d_in[8];
  float* out = (float*)d_out;               // [T,N,C]

  float* Qseq = (float*)d_ws;                 // T*R*C floats (512 KB)
  float* Ysw  = Qseq + (size_t)T_ * R_ * C_;  // T*N*C floats (8 MB), swizzled

  hipLaunchKernelGGL(grcu_qseq, dim3(2), dim3(256), 0, stream, Q0, Wz, Uz, Wr,
                     Ur, Wh, Uh, Qseq);
  hipLaunchKernelGGL(xq_kernel, dim3(T_ * 32), dim3(256), 0, stream, X, Qseq,
                     Ysw);
  hipLaunchKernelGGL(ay_kernel, dim3(T_ * 32), dim3(256), 0, stream, A, Ysw,
                     out);
}